// ENNMessage_20959440404659
// MI455X (gfx1250) — compile-verified
//
#include <hip/hip_runtime.h>

// ---------------------------------------------------------------------------
// Fused ENN message kernel for MI455X (gfx1250, wave32, WMMA).
//   ax = P @ Wr with P[e] = edges[e,:] (+bias/mask row) outer x[src_e,:]
//   built on the fly in registers; Wr pre-shuffled into d_ws in the exact
//   V_WMMA_F32_16X16X4_F32 B-operand lane layout. Scatter via f32 atomics.
// ---------------------------------------------------------------------------

typedef float v2f __attribute__((ext_vector_type(2)));
typedef float v8f __attribute__((ext_vector_type(8)));

#define B_    4
#define N_    4096
#define E_    8192
#define DE_   16
#define D_    64
#define K_    64
#define JTOT  (DE_ + 1)        // 16 edge features + 1 bias row (carries mask & 1/sqrt(K))
#define TSTEPS (JTOT * 16)     // 272 K-chunks of 4 (K-dim = 17*64 = 1088)
#define WS_FLOATS (4 * TSTEPS * 64)   // 4 N-blocks * 272 steps * 32 lanes * 2 = 69632 floats
#define PAD_VAL (-999.0f)

// ---- zero the output (harness poisons d_out) ------------------------------
__global__ void enn_zero_kernel(float* __restrict__ out, int n) {
    int i = blockIdx.x * blockDim.x + threadIdx.x;
    if (i < n) out[i] = 0.0f;
}

// ---- pre-shuffle W (+bias as j==16) into WMMA B-operand layout ------------
// B tile for (nb, t=j*16+dq): lane L (hi=L>>4) supplies B[K=hi*2+v, N=L&15]
//   = Wb[j, (dq*4 + hi*2 + v)*64 + nb*16 + (L&15)]
__global__ void enn_shuffle_w_kernel(const float* __restrict__ W,
                                     const float* __restrict__ bvec,
                                     float* __restrict__ wsB) {
    int f = blockIdx.x * blockDim.x + threadIdx.x;
    if (f >= WS_FLOATS) return;
    int v    = f & 1;
    int lane = (f >> 1) & 31;
    int rest = f >> 6;
    int t    = rest % TSTEPS;
    int nb   = rest / TSTEPS;
    int j    = t >> 4;
    int dq   = t & 15;
    int hi   = lane >> 4;
    int d    = dq * 4 + hi * 2 + v;
    int k    = nb * 16 + (lane & 15);
    int col  = d * K_ + k;
    wsB[f] = (j == DE_) ? bvec[col] : W[j * (D_ * K_) + col];
}

// ---- fused: gather-x -> (edges x W) -> per-edge vec-mat -> scatter-add ----
__global__ __launch_bounds__(128)
void enn_fused_kernel(const float* __restrict__ x,
                      const float* __restrict__ edges,
                      const int*   __restrict__ pairs,
                      const float* __restrict__ wsB,
                      float*       __restrict__ out) {
    __shared__ float se[16 * JTOT];   // masked & scaled edge rows (+bias entry)
    __shared__ float xin[16 * 64];    // gathered x rows for 16 edge-directions
    __shared__ int   dsts[16];
    __shared__ int   srcs[16];

    const int tid    = threadIdx.x;
    const int blk    = blockIdx.x;
    const int b      = blk >> 10;        // 1024 tiles per batch (2E/16)
    const int tile   = blk & 1023;
    const int e2base = tile * 16;
    const float rsK  = 0.125f;           // 1/sqrt(64)

    if (tid < 16) {
        int e2  = e2base + tid;
        int rev = (e2 >= E_);
        int eo  = rev ? (e2 - E_) : e2;
        int p0  = pairs[((size_t)b * E_ + eo) * 2 + 0];
        int p1  = pairs[((size_t)b * E_ + eo) * 2 + 1];
        srcs[tid] = rev ? p0 : p1;       // pairs2[:, :, 1]
        dsts[tid] = rev ? p1 : p0;       // pairs2[:, :, 0]
    }
    // stage edges: se[m][j] = mask * edges[eo,j]/sqrt(K); se[m][16] = mask/sqrt(K)
    for (int t = tid; t < 16 * JTOT; t += 128) {
        int m  = t / JTOT;
        int j  = t % JTOT;
        int e2 = e2base + m;
        int eo = (e2 >= E_) ? (e2 - E_) : e2;
        const float* erow = edges + ((size_t)b * E_ + eo) * DE_;
        float mask = (erow[0] == PAD_VAL) ? 0.0f : 1.0f;
        float val  = (j == DE_) ? 1.0f : erow[j];
        se[m * JTOT + j] = mask * val * rsK;
    }
    __syncthreads();
    // gather x rows (uses srcs written above)
    for (int i = tid; i < 16 * 64; i += 128) {
        int m = i >> 6;
        int d = i & 63;
        xin[i] = x[((size_t)b * N_ + srcs[m]) * D_ + d];
    }
    __syncthreads();

    const int lane = tid & 31;
    const int wid  = tid >> 5;          // N-block 0..3 (16 output cols each)
    const int m    = lane & 15;         // edge-direction row this lane feeds (A operand)
    const int hi   = lane >> 4;         // K half within the 16x4 A tile

    const v2f*   Bw    = (const v2f*)wsB + (size_t)wid * TSTEPS * 32 + lane;
    const float* xrow  = &xin[m * 64 + hi * 2];
    const float* serow = &se[m * JTOT];

    v8f acc = {0.f, 0.f, 0.f, 0.f, 0.f, 0.f, 0.f, 0.f};

    for (int j = 0; j < JTOT; ++j) {
        float sej = serow[j];
#pragma unroll
        for (int dq = 0; dq < 16; ++dq) {
            v2f bv = Bw[(j * 16 + dq) * 32];            // global_load_b64, L2-hot
            v2f xp = *(const v2f*)(xrow + dq * 4);      // ds_load_b64
            v2f av;
            av.x = sej * xp.x;                          // rank-1 A operand on the fly
            av.y = sej * xp.y;
            acc = __builtin_amdgcn_wmma_f32_16x16x4_f32(
                /*neg_a=*/false, av, /*neg_b=*/false, bv,
                /*c_mod=*/(short)0, acc, /*reuse_a=*/false, /*reuse_b=*/false);
        }
    }

    // scatter: C layout lane L, vgpr r -> (M = r + 8*hi, N = L&15)
    const int n   = lane & 15;
    float* outb   = out + (size_t)b * N_ * D_ + wid * 16 + n;
#pragma unroll
    for (int r = 0; r < 8; ++r) {
        int mo = r + hi * 8;
        atomicAdd(outb + (size_t)dsts[mo] * D_, acc[r]);
    }
}

extern "C" void kernel_launch(void* const* d_in, const int* in_sizes, int n_in,
                              void* d_out, int out_size, void* d_ws, size_t ws_size,
                              hipStream_t stream) {
    const float* x     = (const float*)d_in[0];
    const float* edges = (const float*)d_in[1];
    const int*   pairs = (const int*)d_in[2];
    const float* W     = (const float*)d_in[3];
    const float* bvec  = (const float*)d_in[4];
    float* out = (float*)d_out;
    float* wsB = (float*)d_ws;   // needs WS_FLOATS*4 = 278,528 bytes

    enn_zero_kernel<<<(out_size + 255) / 256, 256, 0, stream>>>(out, out_size);
    enn_shuffle_w_kernel<<<(WS_FLOATS + 255) / 256, 256, 0, stream>>>(W, bvec, wsB);
    enn_fused_kernel<<<B_ * 1024, 128, 0, stream>>>(x, edges, pairs, wsB, out);
}